// VQVAE2_67353677136172
// MI455X (gfx1250) — compile-verified
//
#include <hip/hip_runtime.h>
#include <hip/hip_bf16.h>

typedef __attribute__((ext_vector_type(2))) float v2f;
typedef __attribute__((ext_vector_type(4))) float v4f;
typedef __attribute__((ext_vector_type(8))) float v8f;

#define NWAVE 8
#define TROWS 128   // rows per block (8 waves x 16 rows)

__device__ __forceinline__ v8f bc8(float f) {
  v8f r;
#pragma unroll
  for (int i = 0; i < 8; ++i) r[i] = f;
  return r;
}

// One 16x16 output tile: acc += A[16 x 4*KT] * W[4*KT x 16-slice]
// A fragment layout (f32 16x4): lane m = lane%16 holds rows; k = 4t + 2*(lane/16) + j
// B fragment layout (mirror):   lane n = lane%16 holds cols; same k mapping
template <int KT>
__device__ __forceinline__ v8f wmma_acc(const float* A, int lda,
                                        const float* W, int ldw, int coloff,
                                        v8f acc, int m, int n, int h) {
#pragma unroll
  for (int t = 0; t < KT; ++t) {
    const int k = 4 * t + 2 * h;
    v2f a, b;
    a[0] = A[m * lda + k];
    a[1] = A[m * lda + k + 1];
    b[0] = W[k * ldw + coloff + n];
    b[1] = W[(k + 1) * ldw + coloff + n];
    acc = __builtin_amdgcn_wmma_f32_16x16x4_f32(false, a, false, b,
                                                (short)0, acc, false, false);
  }
  return acc;
}

// D-fragment (C/D layout: row m = v + 8*(lane/16), col n = lane%16) -> LDS tile
__device__ __forceinline__ void dtile_to_lds(float* B, int coloff, v8f d,
                                             int n, int h, bool relu) {
#pragma unroll
  for (int v = 0; v < 8; ++v) {
    float val = d[v];
    if (relu) val = fmaxf(val, 0.0f);
    B[(v + 8 * h) * 68 + coloff + n] = val;
  }
}

// D-fragment -> global (ld = 64)
__device__ __forceinline__ void dtile_to_global64(float* g, long grow, int coloff,
                                                  v8f d, int n, int h) {
#pragma unroll
  for (int v = 0; v < 8; ++v)
    g[(grow + v + 8 * h) * 64 + coloff + n] = d[v];
}

// whole 16x64 LDS tile -> global row-major [16][64], coalesced float4 bursts
__device__ __forceinline__ void tile64_to_global(const float* B, float* g,
                                                 long grow, int lane) {
#pragma unroll
  for (int t = 0; t < 8; ++t) {
    int f = t * 32 + lane;          // float4 slot 0..255
    int row = f >> 4;
    int c4 = (f & 15) << 2;
    v4f val = *(const v4f*)(B + row * 68 + c4);
    *(v4f*)(g + (grow + row) * 64 + c4) = val;
  }
}

__global__ __launch_bounds__(256) void vqvae_fwd(
    const float* __restrict__ x,
    const float* __restrict__ enc_w1, const float* __restrict__ enc_b1,
    const float* __restrict__ enc_wm, const float* __restrict__ enc_bm,
    const float* __restrict__ enc_wv, const float* __restrict__ enc_bv,
    const float* __restrict__ codebook,
    const float* __restrict__ res_w1, const float* __restrict__ res_b1,
    const float* __restrict__ res_w2, const float* __restrict__ res_b2,
    const float* __restrict__ dec_w1, const float* __restrict__ dec_b1,
    const float* __restrict__ dec_w2, const float* __restrict__ dec_b2,
    float* __restrict__ out_dec, float* __restrict__ out_mean,
    float* __restrict__ out_logvar, float* __restrict__ out_quant,
    float* __restrict__ out_idx) {
  // ---- weights (transposed, padded strides for bank-conflict-free frags) ----
  __shared__ float w1T[28 * 16], sb1[16];          // enc_w1^T  [K=28][N=16]
  __shared__ float wmT[16 * 68], sbm[64];          // enc_wm^T  [16][64] s=68
  __shared__ float wvT[16 * 68], sbv[64];          // enc_wv^T
  __shared__ float cbT[64 * 16], cbn[16];          // codebook^T [64][16], norms
  __shared__ float cb[16 * 68];                    // codebook rows (gather) s=68
  __shared__ float rw1T[64 * 68], srb1[64];        // res_w1^T [64][64] s=68
  __shared__ float rw2T[64 * 68], srb2[64];
  __shared__ float dw1T[64 * 16], sdb1[16];        // dec_w1^T [64][16]
  __shared__ float dw2T[16 * 36], sdb2[32];        // dec_w2^T [16][28->32] s=36
  __shared__ float xt[TROWS * 28];                 // staged x tile
  __shared__ float bufA[NWAVE][16 * 68];           // per-wave activation ping
  __shared__ float bufB[NWAVE][16 * 68];           // per-wave activation pong
  __shared__ int idxs[NWAVE][16];

  const int tid = threadIdx.x;

  for (int i = tid; i < 28 * 16; i += 256) {
    int k = i >> 4, n = i & 15;
    w1T[k * 16 + n] = enc_w1[n * 28 + k];
  }
  if (tid < 16) { sb1[tid] = enc_b1[tid]; sdb1[tid] = dec_b1[tid]; }
  for (int i = tid; i < 16 * 64; i += 256) {
    int k = i >> 6, n = i & 63;
    wmT[k * 68 + n] = enc_wm[n * 16 + k];
    wvT[k * 68 + n] = enc_wv[n * 16 + k];
  }
  if (tid < 64) {
    sbm[tid] = enc_bm[tid]; sbv[tid] = enc_bv[tid];
    srb1[tid] = res_b1[tid]; srb2[tid] = res_b2[tid];
  }
  for (int i = tid; i < 64 * 16; i += 256) {
    int k = i >> 4, n = i & 15;
    cbT[k * 16 + n] = (n < 10) ? codebook[n * 64 + k] : 0.0f;
    dw1T[k * 16 + n] = dec_w1[n * 64 + k];
  }
  if (tid < 16) {
    float s = 0.0f;
    if (tid < 10)
      for (int k = 0; k < 64; ++k) { float c = codebook[tid * 64 + k]; s += c * c; }
    cbn[tid] = (tid < 10) ? s : 3.0e38f;
  }
  for (int i = tid; i < 16 * 64; i += 256) {
    int r = i >> 6, c = i & 63;
    cb[r * 68 + c] = (r < 10) ? codebook[r * 64 + c] : 0.0f;
  }
  for (int i = tid; i < 64 * 64; i += 256) {
    int k = i >> 6, n = i & 63;
    rw1T[k * 68 + n] = res_w1[n * 64 + k];
    rw2T[k * 68 + n] = res_w2[n * 64 + k];
  }
  for (int i = tid; i < 16 * 32; i += 256) {
    int k = i >> 5, n = i & 31;
    dw2T[k * 36 + n] = (n < 28) ? dec_w2[n * 16 + k] : 0.0f;
  }
  if (tid < 32) sdb2[tid] = (tid < 28) ? dec_b2[tid] : 0.0f;

  // ---- stage x rows for this block (coalesced) ----
  const long tilebase = (long)blockIdx.x * TROWS;
  for (int i = tid; i < TROWS * 28; i += 256) xt[i] = x[tilebase * 28 + i];
  __syncthreads();

  const int lane = tid & 31;
  const int wave = tid >> 5;
  const int m = lane & 15;   // A-frag row / D-frag col
  const int n = m;           // B/D-frag col index
  const int h = lane >> 4;   // half-wave
  const long grow = tilebase + (long)wave * 16;
  float* bA = bufA[wave];
  float* bB = bufB[wave];

  // 1) h = relu(x @ enc_w1^T + b1)   [16x16] -> bufB
  {
    v8f acc = bc8(sb1[n]);
    acc = wmma_acc<7>(xt + wave * 16 * 28, 28, w1T, 16, 0, acc, m, n, h);
    dtile_to_lds(bB, 0, acc, n, h, true);
  }
  __syncthreads();

  // 2) enc_mean = h @ enc_wm^T + bm  [16x64] -> bufA + global
#pragma unroll
  for (int nt = 0; nt < 4; ++nt) {
    v8f acc = bc8(sbm[nt * 16 + n]);
    acc = wmma_acc<4>(bB, 68, wmT, 68, nt * 16, acc, m, n, h);
    dtile_to_lds(bA, nt * 16, acc, n, h, false);
  }
  __syncthreads();
  tile64_to_global(bA, out_mean, grow, lane);

  // 3) enc_logvar = h @ enc_wv^T + bv -> global only
#pragma unroll
  for (int nt = 0; nt < 4; ++nt) {
    v8f acc = bc8(sbv[nt * 16 + n]);
    acc = wmma_acc<4>(bB, 68, wvT, 68, nt * 16, acc, m, n, h);
    dtile_to_global64(out_logvar, grow, nt * 16, acc, n, h);
  }
  __syncthreads();

  // 4) VQ distances: d[m][j] = ||e_j||^2 - 2 * mean_m . e_j  -> bufB[16][16]
  {
    v8f acc = bc8(0.0f);
    acc = wmma_acc<16>(bA, 68, cbT, 16, 0, acc, m, n, h);
    float cn = cbn[n];
    v8f dv;
#pragma unroll
    for (int v = 0; v < 8; ++v) dv[v] = cn - 2.0f * acc[v];
    dtile_to_lds(bB, 0, dv, n, h, false);
  }
  __syncthreads();

  // 5) argmin over 10 codes, emit indices
  if (lane < 16) {
    int best = 0;
    float bval = bB[lane * 68];
    for (int j = 1; j < 10; ++j) {
      float d = bB[lane * 68 + j];
      if (d < bval) { bval = d; best = j; }
    }
    idxs[wave][lane] = best;
    out_idx[grow + lane] = (float)best;
  }
  __syncthreads();

  // 6) gather quantized rows -> bufA, write to global
  {
    int mr = lane >> 1;
    int hf = lane & 1;
    int ci = idxs[wave][mr];
#pragma unroll
    for (int t = 0; t < 8; ++t) {
      int col = hf * 32 + t * 4;
      *(v4f*)(bA + mr * 68 + col) = *(const v4f*)(cb + ci * 68 + col);
    }
  }
  __syncthreads();
  tile64_to_global(bA, out_quant, grow, lane);

  // 7) r = relu(q @ res_w1^T + b1)  -> bufB
#pragma unroll
  for (int nt = 0; nt < 4; ++nt) {
    v8f acc = bc8(srb1[nt * 16 + n]);
    acc = wmma_acc<16>(bA, 68, rw1T, 68, nt * 16, acc, m, n, h);
    dtile_to_lds(bB, nt * 16, acc, n, h, true);
  }
  __syncthreads();

  // 8) qres = r @ res_w2^T + b2 + q  -> bufA (per-tile in-place over q)
#pragma unroll
  for (int nt = 0; nt < 4; ++nt) {
    v8f acc;
#pragma unroll
    for (int v = 0; v < 8; ++v)
      acc[v] = srb2[nt * 16 + n] + bA[(v + 8 * h) * 68 + nt * 16 + n];
    acc = wmma_acc<16>(bB, 68, rw2T, 68, nt * 16, acc, m, n, h);
    dtile_to_lds(bA, nt * 16, acc, n, h, false);
  }
  __syncthreads();

  // 9) d = relu(qres @ dec_w1^T + b1)  [16x16] -> bufB
  {
    v8f acc = bc8(sdb1[n]);
    acc = wmma_acc<16>(bA, 68, dw1T, 16, 0, acc, m, n, h);
    dtile_to_lds(bB, 0, acc, n, h, true);
  }
  __syncthreads();

  // 10) out = sigmoid(d @ dec_w2^T + b2)  [16x28] -> global (masked)
#pragma unroll
  for (int nt = 0; nt < 2; ++nt) {
    v8f acc = bc8(sdb2[nt * 16 + n]);
    acc = wmma_acc<4>(bB, 68, dw2T, 36, nt * 16, acc, m, n, h);
    int col = nt * 16 + n;
    if (col < 28) {
#pragma unroll
      for (int v = 0; v < 8; ++v) {
        float s = 1.0f / (1.0f + __expf(-acc[v]));
        out_dec[(grow + v + 8 * h) * 28 + col] = s;
      }
    }
  }
}

extern "C" void kernel_launch(void* const* d_in, const int* in_sizes, int n_in,
                              void* d_out, int out_size, void* d_ws, size_t ws_size,
                              hipStream_t stream) {
  (void)n_in; (void)out_size; (void)d_ws; (void)ws_size;
  const float* x       = (const float*)d_in[0];
  const float* enc_w1  = (const float*)d_in[1];
  const float* enc_b1  = (const float*)d_in[2];
  const float* enc_wm  = (const float*)d_in[3];
  const float* enc_bm  = (const float*)d_in[4];
  const float* enc_wv  = (const float*)d_in[5];
  const float* enc_bv  = (const float*)d_in[6];
  const float* cbk     = (const float*)d_in[7];
  const float* res_w1  = (const float*)d_in[8];
  const float* res_b1  = (const float*)d_in[9];
  const float* res_w2  = (const float*)d_in[10];
  const float* res_b2  = (const float*)d_in[11];
  const float* dec_w1  = (const float*)d_in[12];
  const float* dec_b1  = (const float*)d_in[13];
  const float* dec_w2  = (const float*)d_in[14];
  const float* dec_b2  = (const float*)d_in[15];

  const long B = (long)in_sizes[0] / 28;
  float* out = (float*)d_out;
  float* out_dec    = out;                 // [B,28]
  float* out_mean   = out + B * 28;        // [B,64]
  float* out_logvar = out + B * 92;        // [B,64]
  float* out_quant  = out + B * 156;       // [B,64]
  float* out_idx    = out + B * 220;       // [B]

  const int nblocks = (int)(B / TROWS);    // B = 1048576 -> 8192, exact
  hipLaunchKernelGGL(vqvae_fwd, dim3(nblocks), dim3(256), 0, stream,
                     x, enc_w1, enc_b1, enc_wm, enc_bm, enc_wv, enc_bv, cbk,
                     res_w1, res_b1, res_w2, res_b2, dec_w1, dec_b1, dec_w2,
                     dec_b2, out_dec, out_mean, out_logvar, out_quant, out_idx);
}